// GATEncoder_67808943669806
// MI455X (gfx1250) — compile-verified
//
#include <hip/hip_runtime.h>
#include <stdint.h>

// ---- problem constants (from reference) ----
#define N_NODES 100000
#define N_EDGES 1600000
#define IN_CH   128
#define HEADS   4
#define OUT_CH  32
#define HC      128            // HEADS*OUT_CH
#define NEG_SLOPE 0.2f
#define EPS_F 1e-16f

// LDS layout for staged weights: 128 data dwords + 8 pad dwords per row
// (TDM pad_interval=128 dwords, pad_amount=8 dwords) -> 2-row offset is
// 272 dwords == 16 mod 64 banks => conflict-free half-wave access.
#define WSTRIDE   136
#define WBYTES    (128 * WSTRIDE * 4)     // 69632 bytes per matrix

typedef __attribute__((ext_vector_type(2))) float v2f;
typedef __attribute__((ext_vector_type(4))) float v4f;
typedef __attribute__((ext_vector_type(8))) float v8f;
typedef __attribute__((ext_vector_type(4))) unsigned int v4u;
typedef __attribute__((ext_vector_type(8))) int v8i;
typedef __attribute__((ext_vector_type(4))) int v4i;

// -------------------------------------------------------------------------
// float atomic max via sign-split int atomics (works with -inf init)
// -------------------------------------------------------------------------
__device__ inline void atomicMaxFloat(float* addr, float val) {
  if (!(val < 0.0f)) {
    atomicMax((int*)addr, __float_as_int(val));
  } else {
    atomicMin((unsigned int*)addr, __float_as_uint(val));
  }
}

// -------------------------------------------------------------------------
// TDM: DMA one 128x128 f32 matrix (row-major, stride 128) from global into
// LDS at byte offset lds_off, inserting 8 pad dwords every 128 dwords so the
// LDS row stride becomes WSTRIDE dwords. D# per CDNA5 ISA 8.3/8.4.
// This toolchain exposes the 6-arg builtin:
//   (uint32x4 g0, int32x8 g1, int32x4 g2, int32x4 g3, int32x8 pad, i32 cpol)
// -------------------------------------------------------------------------
__device__ inline void tdm_load_w(const float* gptr, unsigned lds_off) {
  unsigned long long ga = (unsigned long long)(uintptr_t)gptr;

  v4u g0;
  g0[0] = 1u;                                       // count=1, normal mode
  g0[1] = lds_off;                                  // lds_addr (bytes)
  g0[2] = (unsigned)(ga & 0xFFFFFFFFu);             // global_addr[31:0]
  g0[3] = (unsigned)((ga >> 32) & 0x01FFFFFFu)      // global_addr[56:32]
        | (2u << 30);                               // type=2 ("image")

  v8i g1;
  g1[0] = (2 << 16)        // data_size=2 -> 4 bytes; workgroup_mask=0
        | (1 << 20)        // pad_enable
        | (6 << 22)        // pad_interval: 128 dwords
        | (7 << 25);       // pad_amount:   8 dwords
  g1[1] = (128 << 16);     // atomic_barrier_addr=0 ; tensor_dim0[15:0]=128
  g1[2] = (128 << 16);     // tensor_dim0[31:16]=0  ; tensor_dim1[15:0]=128
  g1[3] = (128 << 16);     // tensor_dim1[31:16]=0  ; tile_dim0=128
  g1[4] = 128;             // tile_dim1=128 ; tile_dim2=0 (2-D tile)
  g1[5] = 128;             // tensor_dim0_stride[31:0]=128 elements
  g1[6] = 0;               // stride0[47:32]=0 ; tensor_dim1_stride lo=0
  g1[7] = 0;               // tensor_dim1_stride hi=0 (unused for 2-D)

  v4i z4 = {0, 0, 0, 0};           // groups 2/3 unused (<=2-D tensor)
  v8i z8 = {0, 0, 0, 0, 0, 0, 0, 0};
  __builtin_amdgcn_tensor_load_to_lds(g0, g1, z4, z4, z8, /*cpol=*/0);
}

// -------------------------------------------------------------------------
// Kernel 0: init out with bias, m = -inf, denom = 0
// -------------------------------------------------------------------------
__global__ void gat_init_kernel(float* __restrict__ out,
                                const float* __restrict__ bias,
                                float* __restrict__ m,
                                float* __restrict__ denom) {
  long long i = (long long)blockIdx.x * blockDim.x + threadIdx.x;
  if (i < (long long)N_NODES * HC) {
    out[i] = bias[(int)(i & (HC - 1))];
  }
  if (i < (long long)N_NODES * HEADS) {
    m[i] = -__builtin_inff();
    denom[i] = 0.0f;
  }
}

// -------------------------------------------------------------------------
// Kernel 1: x_l = x @ W_l, x_r = x @ W_r  via V_WMMA_F32_16X16X4_F32
// W_l / W_r are staged into LDS once per block via the Tensor Data Mover
// (async DMA, TENSORcnt), then the K-loop reads B from LDS while A streams
// from global. Block = 256 threads = 8 waves, 32 rows per block:
//   wave >> 2       : row-tile within block (0/1)
//   (wave >> 1) & 1 : which weight matrix (W_l / W_r)
//   wave & 1        : which 64-column half -> 4 output tiles of 16 cols
// -------------------------------------------------------------------------
__global__ __launch_bounds__(256)
void gat_gemm_kernel(const float* __restrict__ x,
                     const float* __restrict__ Wl,
                     const float* __restrict__ Wr,
                     float* __restrict__ xl,
                     float* __restrict__ xr) {
  extern __shared__ float ldsW[];   // 2 * 128 * WSTRIDE floats

  const int lane  = threadIdx.x & 31;
  const int wave  = threadIdx.x >> 5;                 // 0..7
  const int rowBase = (blockIdx.x * 2 + (wave >> 2)) * 16;
  const bool isR  = (wave >> 1) & 1;
  float* __restrict__ dst = isR ? xr : xl;
  const int colBase = (wave & 1) * 64;

  // --- stage both weight matrices into LDS via TDM (wave 0 only; the DMA
  //     is a wave-level op, EXEC-ignored). Wave 0 waits for completion
  //     BEFORE the barrier so all waves observe the data after it. ---
  if (wave == 0) {
    tdm_load_w(Wl, 0);
    tdm_load_w(Wr, WBYTES);
    __builtin_amdgcn_s_wait_tensorcnt(0);
  }
  __syncthreads();

  const float* __restrict__ Wlds = ldsW + (isR ? (WBYTES / 4) : 0);

  const int r     = lane & 15;     // row (A) / col (B,D) within tile
  const int khalf = lane >> 4;     // 0 or 1 -> K offset 2*khalf

  v8f acc[4] = {};                 // 4 column tiles of 16x16 f32

  const float* __restrict__ aptr =
      x + (long long)(rowBase + r) * IN_CH + khalf * 2;

  for (int kk = 0; kk < IN_CH; kk += 4) {
    v2f a = *(const v2f*)(aptr + kk);        // A: K = kk+2*khalf, +1
    const int krow = kk + khalf * 2;
#pragma unroll
    for (int t = 0; t < 4; ++t) {
      const int col = colBase + t * 16 + r;
      v2f b;
      b.x = Wlds[krow * WSTRIDE + col];
      b.y = Wlds[(krow + 1) * WSTRIDE + col];
      acc[t] = __builtin_amdgcn_wmma_f32_16x16x4_f32(
          /*neg_a=*/false, a, /*neg_b=*/false, b,
          /*c_mod=*/(short)0, acc[t], /*reuse_a=*/false, /*reuse_b=*/false);
    }
  }

#pragma unroll
  for (int t = 0; t < 4; ++t) {
    const int col = colBase + t * 16 + r;
#pragma unroll
    for (int v = 0; v < 8; ++v) {
      const int row = rowBase + v + 8 * khalf;
      dst[(long long)row * HC + col] = acc[t][v];
    }
  }
}

// -------------------------------------------------------------------------
// Kernel 2: per-edge attention logits + segment max (wave per edge)
// lane handles 4 channels [4*lane, 4*lane+4) -> head = lane>>3
// -------------------------------------------------------------------------
__global__ __launch_bounds__(256)
void gat_logits_kernel(const long long* __restrict__ src,
                       const long long* __restrict__ dst,
                       const float* __restrict__ xl,
                       const float* __restrict__ xr,
                       const float* __restrict__ att,
                       float* __restrict__ e,
                       float* __restrict__ m) {
  const int lane = threadIdx.x & 31;
  const long long eid = (long long)blockIdx.x * 8 + (threadIdx.x >> 5);
  if (eid >= N_EDGES) return;
  const int s = (int)src[eid];
  const int d = (int)dst[eid];

  v4f a  = *(const v4f*)(xl + (long long)s * HC + lane * 4);
  v4f b  = *(const v4f*)(xr + (long long)d * HC + lane * 4);
  v4f av = *(const v4f*)(att + lane * 4);

  float sum = 0.0f;
#pragma unroll
  for (int i = 0; i < 4; ++i) {
    const float h  = a[i] + b[i];
    const float lr = (h > 0.0f) ? h : h * NEG_SLOPE;
    sum = __builtin_fmaf(lr, av[i], sum);
  }
  // reduce groups of 8 lanes (one head each); only lanes 0,8,16,24 consume
  sum += __shfl_down(sum, 4, 32);
  sum += __shfl_down(sum, 2, 32);
  sum += __shfl_down(sum, 1, 32);
  if ((lane & 7) == 0) {
    const int h = lane >> 3;
    e[eid * HEADS + h] = sum;
    atomicMaxFloat(&m[(long long)d * HEADS + h], sum);
  }
}

// -------------------------------------------------------------------------
// Kernel 3: e_exp = exp(e - m[dst]) (in place) + segment sum into denom
// -------------------------------------------------------------------------
__global__ void gat_expsum_kernel(const long long* __restrict__ dst,
                                  float* __restrict__ e,
                                  const float* __restrict__ m,
                                  float* __restrict__ denom) {
  const long long i = (long long)blockIdx.x * blockDim.x + threadIdx.x;
  if (i >= (long long)N_EDGES * HEADS) return;
  const long long eid = i >> 2;    // HEADS == 4
  const int h = (int)(i & 3);
  const int d = (int)dst[eid];
  const float ee = __expf(e[i] - m[(long long)d * HEADS + h]);
  e[i] = ee;
  atomicAdd(&denom[(long long)d * HEADS + h], ee);
}

// -------------------------------------------------------------------------
// Kernel 4: out[dst] += x_l[src] * (e_exp / (denom[dst] + eps))  (wave/edge)
// -------------------------------------------------------------------------
__global__ __launch_bounds__(256)
void gat_aggregate_kernel(const long long* __restrict__ src,
                          const long long* __restrict__ dst,
                          const float* __restrict__ xl,
                          const float* __restrict__ ee,
                          const float* __restrict__ denom,
                          float* __restrict__ out) {
  const int lane = threadIdx.x & 31;
  const long long eid = (long long)blockIdx.x * 8 + (threadIdx.x >> 5);
  if (eid >= N_EDGES) return;
  const int s = (int)src[eid];
  const int d = (int)dst[eid];
  const int h = lane >> 3;

  const float alpha = ee[eid * HEADS + h] /
                      (denom[(long long)d * HEADS + h] + EPS_F);

  v4f a = *(const v4f*)(xl + (long long)s * HC + lane * 4);
  float* __restrict__ o = out + (long long)d * HC + lane * 4;
  atomicAdd(o + 0, a[0] * alpha);
  atomicAdd(o + 1, a[1] * alpha);
  atomicAdd(o + 2, a[2] * alpha);
  atomicAdd(o + 3, a[3] * alpha);
}

// -------------------------------------------------------------------------
// launcher
// inputs: 0=x [N,128] f32, 1=edge_index [2,E] i64, 2=W_l [128,128] f32,
//         3=W_r [128,128] f32, 4=att [4,32] f32, 5=bias [128] f32
// -------------------------------------------------------------------------
extern "C" void kernel_launch(void* const* d_in, const int* in_sizes, int n_in,
                              void* d_out, int out_size, void* d_ws, size_t ws_size,
                              hipStream_t stream) {
  const float* x            = (const float*)d_in[0];
  const long long* ei       = (const long long*)d_in[1];
  const float* Wl           = (const float*)d_in[2];
  const float* Wr           = (const float*)d_in[3];
  const float* att          = (const float*)d_in[4];
  const float* bias         = (const float*)d_in[5];
  float* out                = (float*)d_out;

  const long long* src  = ei;            // edge_index[0, :]
  const long long* dstI = ei + N_EDGES;  // edge_index[1, :]

  // workspace carve-up (all f32):
  float* xl    = (float*)d_ws;                       // N*HC
  float* xr    = xl + (size_t)N_NODES * HC;          // N*HC
  float* e     = xr + (size_t)N_NODES * HC;          // E*HEADS (logits -> e_exp)
  float* m     = e  + (size_t)N_EDGES * HEADS;       // N*HEADS
  float* denom = m  + (size_t)N_NODES * HEADS;       // N*HEADS

  // 0) init out=bias, m=-inf, denom=0
  {
    long long total = (long long)N_NODES * HC;
    int blocks = (int)((total + 255) / 256);
    gat_init_kernel<<<blocks, 256, 0, stream>>>(out, bias, m, denom);
  }
  // 1) x_l / x_r transform via WMMA, weights staged by TDM (N % 32 == 0)
  gat_gemm_kernel<<<N_NODES / 32, 256, 2 * WBYTES, stream>>>(x, Wl, Wr, xl, xr);
  // 2) logits + segment max (8 edges per block)
  gat_logits_kernel<<<(N_EDGES + 7) / 8, 256, 0, stream>>>(src, dstI, xl, xr, att, e, m);
  // 3) exp + segment sum
  {
    long long total = (long long)N_EDGES * HEADS;
    int blocks = (int)((total + 255) / 256);
    gat_expsum_kernel<<<blocks, 256, 0, stream>>>(dstI, e, m, denom);
  }
  // 4) weighted aggregation
  gat_aggregate_kernel<<<(N_EDGES + 7) / 8, 256, 0, stream>>>(src, dstI, xl, e, denom, out);
}